// UpsampleSparse_22222160789823
// MI455X (gfx1250) — compile-verified
//
#include <hip/hip_runtime.h>
#include <hip/hip_bf16.h>

typedef unsigned int u32;

#define PKEYS        (1u << 22)                    // 2*128*128*128 parent-key space
#define SCAN_THREADS 256
#define SCAN_ELEMS   16                            // elements per thread
#define SCAN_CHUNK   (SCAN_THREADS * SCAN_ELEMS)   // 4096
#define SCAN_BLOCKS  (PKEYS / SCAN_CHUNK)          // 1024

// ---------------------------------------------------------------- zero occ
__global__ void us_zero(u32* __restrict__ occ) {
    u32 i = blockIdx.x * blockDim.x + threadIdx.x;   // PKEYS/4 uint4 stores
    ((uint4*)occ)[i] = make_uint4(0u, 0u, 0u, 0u);
}

// ------------------------------------------------------------- scatter 1s
__global__ void us_scatter(const int* __restrict__ coords, u32* __restrict__ occ, int n) {
    int i = blockIdx.x * blockDim.x + threadIdx.x;
    if (i < n) {
        int4 c = ((const int4*)coords)[i];           // (b,z,y,x)
        u32 pkey = (((((u32)c.x << 7) | (u32)c.y) << 7) | (u32)c.z) << 7 | (u32)c.w;
        occ[pkey] = 1u;                              // coords unique -> no atomics
    }
}

// ------------------------------------------------- per-block sums of occ
__global__ void us_blocksum(const u32* __restrict__ occ, u32* __restrict__ bsum) {
    __shared__ u32 sh[SCAN_THREADS];
    u32 t = threadIdx.x;
    u32 base = blockIdx.x * SCAN_CHUNK;
    u32 s = 0;
    #pragma unroll
    for (int k = 0; k < SCAN_ELEMS; ++k) s += occ[base + t + (u32)k * SCAN_THREADS];
    sh[t] = s;
    __syncthreads();
    for (int off = SCAN_THREADS / 2; off > 0; off >>= 1) {
        if (t < (u32)off) sh[t] += sh[t + off];
        __syncthreads();
    }
    if (t == 0) bsum[blockIdx.x] = sh[0];
}

// ------------------------------------ exclusive scan of block sums (1 CTA)
__global__ void us_scanbsum(u32* __restrict__ bsum) {
    __shared__ u32 sh[SCAN_BLOCKS];
    for (u32 i = threadIdx.x; i < SCAN_BLOCKS; i += blockDim.x) sh[i] = bsum[i];
    __syncthreads();
    if (threadIdx.x == 0) {
        u32 run = 0;
        for (int i = 0; i < SCAN_BLOCKS; ++i) { u32 v = sh[i]; sh[i] = run; run += v; }
    }
    __syncthreads();
    for (u32 i = threadIdx.x; i < SCAN_BLOCKS; i += blockDim.x) bsum[i] = sh[i];
}

// ------------------- in-place exclusive scan: occ[] becomes S[] (prefix counts)
__global__ void us_scanfinal(u32* __restrict__ occ, const u32* __restrict__ bsum) {
    __shared__ u32 sh[SCAN_THREADS];
    u32 t = threadIdx.x;
    u32 base = blockIdx.x * SCAN_CHUNK + t * SCAN_ELEMS;
    uint4* p = (uint4*)(occ + base);
    uint4 a0 = p[0], a1 = p[1], a2 = p[2], a3 = p[3];
    u32 v[SCAN_ELEMS] = { a0.x, a0.y, a0.z, a0.w, a1.x, a1.y, a1.z, a1.w,
                          a2.x, a2.y, a2.z, a2.w, a3.x, a3.y, a3.z, a3.w };
    u32 sum = 0;
    #pragma unroll
    for (int k = 0; k < SCAN_ELEMS; ++k) { u32 x = v[k]; v[k] = sum; sum += x; }
    sh[t] = sum;
    __syncthreads();
    // Hillis-Steele inclusive scan over thread sums
    for (int off = 1; off < SCAN_THREADS; off <<= 1) {
        u32 w = (t >= (u32)off) ? sh[t - off] : 0u;
        __syncthreads();
        sh[t] += w;
        __syncthreads();
    }
    u32 prefix = sh[t] - sum + bsum[blockIdx.x];     // exclusive within block + block offset
    #pragma unroll
    for (int k = 0; k < SCAN_ELEMS; ++k) v[k] += prefix;
    p[0] = make_uint4(v[0], v[1], v[2], v[3]);
    p[1] = make_uint4(v[4], v[5], v[6], v[7]);
    p[2] = make_uint4(v[8], v[9], v[10], v[11]);
    p[3] = make_uint4(v[12], v[13], v[14], v[15]);
}

// -------- emit: 16 lanes per child (64 floats = 16×float4), rank in closed form
__global__ void __launch_bounds__(256)
us_emit(const float* __restrict__ feats, const int* __restrict__ coords,
        const u32* __restrict__ S, float* __restrict__ of, int* __restrict__ oc, int n) {
    int child = blockIdx.x * 16 + ((int)threadIdx.x >> 4);
    int lane  = (int)threadIdx.x & 15;
    int M = n * 8;
    if (child >= M) return;
    int i = child >> 3;
    int o = child & 7;
    int dz = (o >> 2) & 1, dy = (o >> 1) & 1, dx = o & 1;

    const float4* src = (const float4*)(feats + (size_t)i * 64);
    __builtin_prefetch(src + lane, 0, 0);            // global_prefetch_b8

    int4 c = ((const int4*)coords)[i];               // (b,z,y,x)
    u32 bz   = ((u32)c.x << 7) | (u32)c.y;
    u32 bzy  = (bz << 7) | (u32)c.z;
    u32 pkey = (bzy << 7) | (u32)c.w;

    u32 s_bz   = S[bz << 14];
    u32 i_bz1  = (bz + 1u) << 14;
    u32 s_bz1  = (i_bz1 >= PKEYS) ? (u32)n : S[i_bz1];
    u32 s_bzy  = S[bzy << 7];
    u32 i_bzy1 = (bzy + 1u) << 7;
    u32 s_bzy1 = (i_bzy1 >= PKEYS) ? (u32)n : S[i_bzy1];
    u32 s_p    = S[pkey];

    u32 A = s_bz;
    u32 B = s_bz1 - s_bz;
    u32 C = s_bzy - s_bz;
    u32 E = s_bzy1 - s_bzy;
    u32 F = s_p - s_bzy;
    u32 rank = 8u * A + 4u * (u32)dz * B + 4u * C + 2u * (u32)dy * E + 2u * F + (u32)dx;

    float4* dst = (float4*)(of + (size_t)rank * 64);
    dst[lane] = src[lane];                           // global_load_b128 / global_store_b128

    if (lane == 0) {
        ((int4*)oc)[rank] = make_int4(c.x, (c.y << 1) | dz, (c.z << 1) | dy, (c.w << 1) | dx);
    }
}

extern "C" void kernel_launch(void* const* d_in, const int* in_sizes, int n_in,
                              void* d_out, int out_size, void* d_ws, size_t ws_size,
                              hipStream_t stream) {
    const float* feats  = (const float*)d_in[0];
    const int*   coords = (const int*)d_in[1];
    // d_in[2] = scale (must be 2; hardcoded as in the reference module)
    int n = in_sizes[1] / 4;                 // N parents
    int M = n * 8;                           // children

    u32* occ  = (u32*)d_ws;                  // PKEYS u32 -> becomes S (prefix counts)
    u32* bsum = occ + PKEYS;                 // SCAN_BLOCKS u32

    float* of = (float*)d_out;               // out_feats: M*64 f32
    int*   oc = (int*)(of + (size_t)M * 64); // out_coords: M*4 i32 (bit-exact region)

    us_zero     <<<PKEYS / 4 / 256, 256, 0, stream>>>(occ);
    us_scatter  <<<(n + 255) / 256, 256, 0, stream>>>(coords, occ, n);
    us_blocksum <<<SCAN_BLOCKS, SCAN_THREADS, 0, stream>>>(occ, bsum);
    us_scanbsum <<<1, 256, 0, stream>>>(bsum);
    us_scanfinal<<<SCAN_BLOCKS, SCAN_THREADS, 0, stream>>>(occ, bsum);
    us_emit     <<<(M + 15) / 16, 256, 0, stream>>>(feats, coords, occ, of, oc, n);
}